// FeaturesLoss_20538533609491
// MI455X (gfx1250) — compile-verified
//
#include <hip/hip_runtime.h>
#include <hip/hip_bf16.h>

typedef __attribute__((ext_vector_type(2))) float v2f;
typedef __attribute__((ext_vector_type(8))) float v8f;

#define N_ROWS 4096
#define KDIM   512
#define NT32   128           // 4096/32 macro-tiles per dimension
#define NTILES 8256          // NT32*(NT32+1)/2 upper-triangular 32x32 macro-tiles
#define NBLK   1032          // NTILES / 8 waves per block

__device__ __forceinline__ float wave_reduce(float v) {
    // wave32 butterfly reduce
    #pragma unroll
    for (int o = 16; o > 0; o >>= 1) v += __shfl_xor(v, o, 32);
    return v;
}

// ---------------- kernel 1: per-row squared norms ----------------
__global__ void fl_prep_sq(const float* __restrict__ X, float* __restrict__ sq) {
    const int lane = threadIdx.x & 31;
    const int wave = threadIdx.x >> 5;
    const int row  = blockIdx.x * 8 + wave;          // 512 blocks * 8 waves = 4096 rows
    const float4* p = (const float4*)(X + (size_t)row * KDIM);
    float s = 0.0f;
    #pragma unroll
    for (int k = 0; k < 4; ++k) {
        float4 v = p[lane + 32 * k];
        s += v.x * v.x + v.y * v.y + v.z * v.z + v.w * v.w;
    }
    s = wave_reduce(s);
    if (lane == 0) sq[row] = s;
}

// epilogue helper: accumulate loss terms for one 16x16 sub-tile
__device__ __forceinline__ void fl_subtile(const v8f& acc, int i0, int j0,
                                           const float* __restrict__ sq,
                                           const int* __restrict__ labels,
                                           float margin, float w,
                                           float& s1, float& s2, float& c1) {
    const int lane  = threadIdx.x & 31;
    const int n     = lane & 15;
    const int mbase = (lane >> 4) * 8;
    const int j     = j0 + n;
    const float sqj = sq[j];
    const int  labj = labels[j];
    #pragma unroll
    for (int v = 0; v < 8; ++v) {
        const int i = i0 + mbase + v;
        float d = sq[i] + sqj - 2.0f * acc[v];
        d = fmaxf(d, 0.0f);
        if (labels[i] == labj) { s1 += w * d; c1 += w; }
        else                   { s2 += w * fmaxf(0.0f, margin - d); }
    }
}

// ---------------- kernel 2: WMMA gram macro-tiles + loss partials ----------------
__global__ void fl_main(const float* __restrict__ X,
                        const int*   __restrict__ labels,
                        const float* __restrict__ marginPtr,
                        const float* __restrict__ sq,
                        float*       __restrict__ partial) {
    const int lane = threadIdx.x & 31;
    const int wave = threadIdx.x >> 5;
    const int t    = blockIdx.x * 8 + wave;          // macro-tile id in upper triangle
    __shared__ float sh[3][8];

    float s1 = 0.0f, s2 = 0.0f, c1 = 0.0f;

    if (t < NTILES) {
        // decode (it, jt), jt >= it, from triangular index (float sqrt + exact fixup)
        float disc = (float)(2 * NT32 + 1) * (float)(2 * NT32 + 1) - 8.0f * (float)t;
        int it = (int)(((float)(2 * NT32 + 1) - sqrtf(disc)) * 0.5f);
        if (it < 0) it = 0; if (it > NT32 - 1) it = NT32 - 1;
        while (((it + 1) * NT32 - ((it + 1) * it) / 2) <= t) ++it;
        while ((it * NT32 - (it * (it - 1)) / 2) > t) --it;
        const int jt = it + (t - (it * NT32 - (it * (it - 1)) / 2));

        const int i0 = it * 32;
        const int j0 = jt * 32;

        // f32 WMMA 16x16x4 fragment mapping (ISA 7.12.2):
        //   A: lane L holds row m = L%16, K pair base 2*(L/16), 2 consecutive K per lane
        //   B (KxN): lane L holds col n = L%16, same K mapping -> identical load pattern
        const int mrow = lane & 15;
        const int kq   = (lane >> 4) * 2;
        const float* Alo = X + (size_t)(i0 + mrow) * KDIM + kq;
        const float* Ahi = Alo + (size_t)16 * KDIM;
        const float* Blo = X + (size_t)(j0 + mrow) * KDIM + kq;
        const float* Bhi = Blo + (size_t)16 * KDIM;

        v8f c00 = {}; v8f c01 = {}; v8f c10 = {}; v8f c11 = {};
        #pragma unroll 2
        for (int kb = 0; kb < KDIM; kb += 4) {
            v2f alo = *(const v2f*)(Alo + kb);
            v2f ahi = *(const v2f*)(Ahi + kb);
            v2f blo = *(const v2f*)(Blo + kb);
            v2f bhi = *(const v2f*)(Bhi + kb);
            c00 = __builtin_amdgcn_wmma_f32_16x16x4_f32(false, alo, false, blo, (short)0, c00, false, false);
            c01 = __builtin_amdgcn_wmma_f32_16x16x4_f32(false, alo, false, bhi, (short)0, c01, false, false);
            c10 = __builtin_amdgcn_wmma_f32_16x16x4_f32(false, ahi, false, blo, (short)0, c10, false, false);
            c11 = __builtin_amdgcn_wmma_f32_16x16x4_f32(false, ahi, false, bhi, (short)0, c11, false, false);
        }

        const float margin = *marginPtr;
        // diagonal macro-tile contains both mirror sub-tiles -> weight 1 everywhere;
        // off-diagonal macro-tile's mirror is skipped -> weight 2 everywhere
        const float w = (it == jt) ? 1.0f : 2.0f;

        fl_subtile(c00, i0,      j0,      sq, labels, margin, w, s1, s2, c1);
        fl_subtile(c01, i0,      j0 + 16, sq, labels, margin, w, s1, s2, c1);
        fl_subtile(c10, i0 + 16, j0,      sq, labels, margin, w, s1, s2, c1);
        fl_subtile(c11, i0 + 16, j0 + 16, sq, labels, margin, w, s1, s2, c1);
    }

    s1 = wave_reduce(s1);
    s2 = wave_reduce(s2);
    c1 = wave_reduce(c1);
    if (lane == 0) { sh[0][wave] = s1; sh[1][wave] = s2; sh[2][wave] = c1; }
    __syncthreads();
    if (threadIdx.x == 0) {
        float t1 = 0.0f, t2 = 0.0f, t3 = 0.0f;
        #pragma unroll
        for (int wv = 0; wv < 8; ++wv) { t1 += sh[0][wv]; t2 += sh[1][wv]; t3 += sh[2][wv]; }
        partial[3 * blockIdx.x + 0] = t1;
        partial[3 * blockIdx.x + 1] = t2;
        partial[3 * blockIdx.x + 2] = t3;
    }
}

// ---------------- kernel 3: deterministic final reduce ----------------
__global__ void fl_finalize(const float* __restrict__ partial, float* __restrict__ out) {
    __shared__ double r1[256], r2[256], r3[256];
    const int tid = threadIdx.x;
    double s1 = 0.0, s2 = 0.0, c1 = 0.0;
    for (int i = tid; i < NBLK; i += 256) {
        s1 += (double)partial[3 * i + 0];
        s2 += (double)partial[3 * i + 1];
        c1 += (double)partial[3 * i + 2];
    }
    r1[tid] = s1; r2[tid] = s2; r3[tid] = c1;
    __syncthreads();
    for (int s = 128; s > 0; s >>= 1) {
        if (tid < s) { r1[tid] += r1[tid + s]; r2[tid] += r2[tid + s]; r3[tid] += r3[tid + s]; }
        __syncthreads();
    }
    if (tid == 0) {
        const double zn1 = r3[0];
        const double zn2 = (double)N_ROWS * (double)N_ROWS - zn1;
        out[0] = (float)(0.5 * (r1[0] / zn1 + r2[0] / zn2));
    }
}

extern "C" void kernel_launch(void* const* d_in, const int* in_sizes, int n_in,
                              void* d_out, int out_size, void* d_ws, size_t ws_size,
                              hipStream_t stream) {
    const float* X      = (const float*)d_in[0];
    const int*   labels = (const int*)d_in[1];    // jax default: int64 request -> int32 array
    const float* margin = (const float*)d_in[2];
    float* out = (float*)d_out;

    float* sq      = (float*)d_ws;                // 4096 floats
    float* partial = sq + N_ROWS;                 // NBLK * 3 floats

    fl_prep_sq<<<N_ROWS / 8, 256, 0, stream>>>(X, sq);
    fl_main<<<NBLK, 256, 0, stream>>>(X, labels, margin, sq, partial);
    fl_finalize<<<1, 256, 0, stream>>>(partial, out);
}